// FC_CPPN_60232621359502
// MI455X (gfx1250) — compile-verified
//
#include <hip/hip_runtime.h>
#include <math.h>

typedef __attribute__((ext_vector_type(2))) float v2f;
typedef __attribute__((ext_vector_type(8))) float v8f;

#define MOTION 8
#define HID 8
#define NOUT 3
#define NLAYERS 3
#define INV_SQRT_2PI 0.3989422804014327f

// cross-lane swap of lanes 0-15 <-> 16-31 (ds_swizzle SWAPX16: xor=0x10, and=0x1f)
__device__ __forceinline__ float swap16(float x) {
    int i = __builtin_bit_cast(int, x);
    i = __builtin_amdgcn_ds_swizzle(i, 0x401F);
    return __builtin_bit_cast(float, i);
}

__device__ __forceinline__ v8f wmma4(float a0, float a1, float b0, float b1, v8f c) {
    v2f a = {a0, a1};
    v2f b = {b0, b1};
    return __builtin_amdgcn_wmma_f32_16x16x4_f32(false, a, false, b, (short)0, c, false, false);
}

__global__ __launch_bounds__(256) void cppn_wmma_kernel(
    const float* __restrict__ x, const float* __restrict__ y,
    const float* __restrict__ r, const float* __restrict__ z,
    const float* __restrict__ W0, const float* __restrict__ b0,
    const float* __restrict__ Wm, const float* __restrict__ bm,
    const float* __restrict__ Wo, const float* __restrict__ bo,
    const int* __restrict__ masks, float* __restrict__ outp, int n)
{
    const int lane = threadIdx.x & 31;
    const int wave = threadIdx.x >> 5;
    const int hl   = lane >> 4;        // half-wave: K offset +2
    const int col  = lane & 15;        // A row / B,D pixel column
    const int p    = blockIdx.x * 128 + wave * 16 + col;
    const int pc   = (p < n) ? p : (n - 1);

    // ---- packed per-layer activation codes (8 x 2-bit), forced wave-uniform ----
    int code[NLAYERS];
    #pragma unroll
    for (int l = 0; l < NLAYERS; ++l) {
        int cd = 0;
        #pragma unroll
        for (int e = 0; e < 8; ++e)            // masks[l, f=e>>1, j=e&1]
            cd |= (e >> 1) << (2 * masks[l * 8 + e]);
        code[l] = __builtin_amdgcn_readfirstlane(cd);
    }

    // ---- branchless A-operand weight fetch: K = 4c + 2*hl + j, M = col ----
    const int rowc = col & 7;                  // clamped row (always valid addr)
    const int rowo = (col > 2) ? 2 : col;      // clamped row for Wo
    float w0a[3][2], wma[2][2], woa[2][2];
    #pragma unroll
    for (int c = 0; c < 3; ++c)
        #pragma unroll
        for (int j = 0; j < 2; ++j) {
            int k  = 4 * c + 2 * hl + j;
            int kc = (k > 10) ? 10 : k;
            float v = W0[rowc * (MOTION + 3) + kc];
            w0a[c][j] = (col < HID && k < MOTION + 3) ? v : 0.0f;
        }
    #pragma unroll
    for (int c = 0; c < 2; ++c)
        #pragma unroll
        for (int j = 0; j < 2; ++j) {
            int k = 4 * c + 2 * hl + j;
            float vm = Wm[rowc * HID + k];
            float vo = Wo[rowo * HID + k];
            wma[c][j] = (col < HID) ? vm : 0.0f;
            woa[c][j] = (col < NOUT) ? vo : 0.0f;
        }

    // ---- hoisted bias C-operands (rows 8-15 / lanes 16-31 are zero padding) ----
    v8f cb0, cbm, cbo;
    #pragma unroll
    for (int i = 0; i < 8; ++i) {
        cb0[i] = hl ? 0.0f : b0[i];
        cbm[i] = hl ? 0.0f : bm[i];
        cbo[i] = (hl == 0 && i < NOUT) ? bo[i] : 0.0f;
    }

    // ---- layer-0 B operands: in[k,p] = [z/10 (k<8), x, y, r, pad] ----
    const float2* zp = (const float2*)(z + (size_t)pc * MOTION);
    float2 z0 = zp[hl];        // k = 2hl, 2hl+1
    float2 z1 = zp[2 + hl];    // k = 4+2hl, 4+2hl+1
    const float zs = 0.1f;
    float b00 = z0.x * zs, b01 = z0.y * zs;
    float b10 = z1.x * zs, b11 = z1.y * zs;
    const float* pxr = hl ? r : x;
    float b20 = pxr[pc];       // k = 8 (x) | 10 (r)
    float b21 = y[pc];         // k = 9 (y) | 11 (A column is zero -> value ignored)

    // ---- layer 0: 3 chained f32 WMMAs (K=4) ----
    v8f acc = wmma4(w0a[0][0], w0a[0][1], b00, b01, cb0);
    acc     = wmma4(w0a[1][0], w0a[1][1], b10, b11, acc);
    acc     = wmma4(w0a[2][0], w0a[2][1], b20, b21, acc);

    float o[8];
    #pragma unroll
    for (int i = 0; i < 8; ++i) o[i] = acc[i];

    // ---- hidden layers: out = (act(Wm*out + bm) + out) * 0.5 ----
    #pragma unroll
    for (int l = 0; l < NLAYERS; ++l) {
        float s2 = swap16(o[2]), s3 = swap16(o[3]);
        float s6 = swap16(o[6]), s7 = swap16(o[7]);
        v8f c = wmma4(wma[0][0], wma[0][1], hl ? s2 : o[0], hl ? s3 : o[1], cbm);
        c     = wmma4(wma[1][0], wma[1][1], hl ? s6 : o[4], hl ? s7 : o[5], c);
        const int cd = code[l];
        #pragma unroll
        for (int i = 0; i < 8; ++i) {
            float t = c[i], av;
            int f = (cd >> (2 * i)) & 3;       // scalar -> uniform branches
            if (f == 0)      av = __sinf(t);
            else if (f == 1) av = INV_SQRT_2PI * __expf(-0.5f * t * t);
            else if (f == 2) {
                float e = __expf(2.0f * t);    // tanh = 1 - 2/(e^{2t}+1)
                av = 1.0f - 2.0f * __builtin_amdgcn_rcpf(e + 1.0f);
            } else           av = t;
            o[i] = (av + o[i]) * 0.5f;
        }
    }

    // ---- output layer: sigmoid(Wo*out + bo), rows 0..2 real ----
    {
        float s2 = swap16(o[2]), s3 = swap16(o[3]);
        float s6 = swap16(o[6]), s7 = swap16(o[7]);
        v8f c = wmma4(woa[0][0], woa[0][1], hl ? s2 : o[0], hl ? s3 : o[1], cbo);
        c     = wmma4(woa[1][0], woa[1][1], hl ? s6 : o[4], hl ? s7 : o[5], c);

        if (hl == 0 && p < n) {
            #pragma unroll
            for (int j = 0; j < NOUT; ++j)
                outp[p * NOUT + j] =
                    __builtin_amdgcn_rcpf(1.0f + __expf(-c[j]));
        }
    }
}

extern "C" void kernel_launch(void* const* d_in, const int* in_sizes, int n_in,
                              void* d_out, int out_size, void* d_ws, size_t ws_size,
                              hipStream_t stream) {
    const float* x  = (const float*)d_in[0];
    const float* y  = (const float*)d_in[1];
    const float* r  = (const float*)d_in[2];
    const float* z  = (const float*)d_in[3];
    const float* W0 = (const float*)d_in[4];
    const float* b0 = (const float*)d_in[5];
    const float* Wm = (const float*)d_in[6];
    const float* bm = (const float*)d_in[7];
    const float* Wo = (const float*)d_in[8];
    const float* bo = (const float*)d_in[9];
    const int* masks = (const int*)d_in[10];
    float* outp = (float*)d_out;
    int n = in_sizes[0];
    int grid = (n + 127) / 128;   // 128 pixels per 256-thread block (8 waves x 16 px)
    cppn_wmma_kernel<<<grid, 256, 0, stream>>>(x, y, r, z, W0, b0, Wm, bm, Wo, bo,
                                               masks, outp, n);
}